// DGCNN_cls_81733227642905
// MI455X (gfx1250) — compile-verified
//
#include <hip/hip_runtime.h>
#include <math.h>
#include <stdint.h>

#define NPTS 4096
#define NB   8
#define KNN  20
#define EPSBN 1e-5f

typedef float v2f __attribute__((ext_vector_type(2)));
typedef float v8f __attribute__((ext_vector_type(8)));

// -------------------------------------------------------------------------
// xx[b,n] = sum_c h[b,c,n]^2
// -------------------------------------------------------------------------
__global__ void sqnorm_kernel(const float* __restrict__ h, int C,
                              float* __restrict__ xx) {
    int t = blockIdx.x * blockDim.x + threadIdx.x;
    if (t >= NB * NPTS) return;
    int b = t / NPTS, n = t % NPTS;
    const float* hb = h + (size_t)b * C * NPTS;
    float s = 0.f;
    for (int c = 0; c < C; ++c) { float v = hb[c * NPTS + n]; s += v * v; }
    xx[t] = s;
}

// -------------------------------------------------------------------------
// kNN: 4 waves / workgroup, 64 rows. Column panels (C x 16) are staged into
// LDS with GLOBAL_LOAD_ASYNC_TO_LDS_B128, double-buffered so the next tile's
// fetch overlaps WMMA + top-k insertion of the current tile. All 4 waves
// share one staged B panel (4x reuse). Inner products via WMMA f32 16x16x4;
// neg_d = 2*inner - xx_i - xx_j; per-row top-20 insertion lists in LDS.
// -------------------------------------------------------------------------
template <int CK>
__global__ __launch_bounds__(128)
void knn_kernel(const float* __restrict__ x, const float* __restrict__ xx,
                int* __restrict__ idx, int C) {
    constexpr int NT    = NPTS / 16;       // 256 column tiles
    constexpr int WORDS = CK * 64;         // floats per stage buffer (CK*4 ch x 16)
    constexpr int NLD   = CK * 16;         // b128 loads per tile
    constexpr int P     = (NLD + 127) / 128; // async instrs per wave per tile

    const int b    = blockIdx.y;
    const int tid  = threadIdx.x;
    const int wave = tid >> 5;
    const int lane = tid & 31;
    const int hf   = lane >> 4;            // half-wave 0/1
    const int lm   = lane & 15;
    const int i0   = blockIdx.x * 64 + wave * 16;
    const float* xb  = x  + (size_t)b * C * NPTS;
    const float* xxb = xx + (size_t)b * NPTS;

    __shared__ float stage[2][WORDS];
    __shared__ float tile[4][16][17];
    __shared__ float tkv[64][KNN];
    __shared__ int   tki[64][KNN];

    if (lane < 16) {
        const int r = wave * 16 + lane;
        #pragma unroll
        for (int k = 0; k < KNN; ++k) { tkv[r][k] = -3.0e38f; tki[r][k] = 0; }
    }
    // zero the channel-pad region of both stage buffers (only when C < CK*4)
    for (int w = tid; w < WORDS; w += 128)
        if (w >= C * 16) { stage[0][w] = 0.f; stage[1][w] = 0.f; }

    // Cache A fragments (rows i0..i0+15, all channels, K padded to 4*CK)
    v2f afr[CK];
    #pragma unroll
    for (int kk = 0; kk < CK; ++kk) {
        int c0 = kk * 4 + hf * 2;
        afr[kk].x = (c0     < C) ? xb[(size_t)(c0    ) * NPTS + i0 + lm] : 0.0f;
        afr[kk].y = (c0 + 1 < C) ? xb[(size_t)(c0 + 1) * NPTS + i0 + lm] : 0.0f;
    }
    float xi[8];
    #pragma unroll
    for (int r = 0; r < 8; ++r) xi[r] = xxb[i0 + hf * 8 + r];

    // async-stage one column panel: quad i covers words 4i..4i+3 (contiguous
    // in n within one channel row since 4 | 16)
    auto issue = [&](int jt, int buf) {
        const int j0 = jt * 16;
        for (int i = tid; i < NLD; i += 128) {
            const int word = 4 * i;
            const int c    = word >> 4;
            const int nn   = word & 15;
            if (c < C) {
                unsigned lds = (unsigned)(uintptr_t)&stage[buf][word];
                const float* gp = xb + (size_t)c * NPTS + j0 + nn;
                asm volatile("global_load_async_to_lds_b128 %0, %1, off"
                             :: "v"(lds), "v"((unsigned long long)(uintptr_t)gp)
                             : "memory");
            }
        }
    };

    issue(0, 0);                            // prologue: tile 0 -> buffer 0

    for (int jt = 0; jt < NT; ++jt) {
        const int cur = jt & 1;
        const int j0  = jt * 16;

        if (jt + 1 < NT) {
            issue(jt + 1, cur ^ 1);         // prefetch next tile
            if constexpr (P >= 2)
                asm volatile("s_wait_asynccnt 0x2" ::: "memory");
            else
                asm volatile("s_wait_asynccnt 0x1" ::: "memory");
        } else {
            asm volatile("s_wait_asynccnt 0x0" ::: "memory");
        }
        __syncthreads();                    // panel resident for all waves

        v8f acc = {0.f,0.f,0.f,0.f,0.f,0.f,0.f,0.f};
        #pragma unroll
        for (int kk = 0; kk < CK; ++kk) {
            int c0 = kk * 4 + hf * 2;
            v2f bfr;
            bfr.x = stage[cur][(c0    ) * 16 + lm];
            bfr.y = stage[cur][(c0 + 1) * 16 + lm];
            acc = __builtin_amdgcn_wmma_f32_16x16x4_f32(
                false, afr[kk], false, bfr, (short)0, acc, false, false);
        }
        float xj = xxb[j0 + lm];
        #pragma unroll
        for (int r = 0; r < 8; ++r)
            tile[wave][hf * 8 + r][lm] = 2.0f * acc[r] - xi[r] - xj;
        __syncthreads();

        if (lane < 16) {
            const int r = wave * 16 + lane;
            for (int jj = 0; jj < 16; ++jj) {
                float v = tile[wave][lane][jj];
                if (v > tkv[r][KNN - 1]) {
                    int p = KNN - 1;
                    while (p > 0 && tkv[r][p - 1] < v) {
                        tkv[r][p] = tkv[r][p - 1];
                        tki[r][p] = tki[r][p - 1];
                        --p;
                    }
                    tkv[r][p] = v; tki[r][p] = j0 + jj;
                }
            }
        }
        __syncthreads();                    // all reads of stage[cur] done
    }

    if (lane < 16) {
        const int r = wave * 16 + lane;
        int* orow = idx + ((size_t)b * NPTS + i0 + lane) * KNN;
        for (int k = 0; k < KNN; ++k) orow[k] = tki[r][k];
    }
}

// -------------------------------------------------------------------------
// g[b,o,n] = sum_c W[o,c] * h[b,c,n]   (M=64 via 4 row-tiles, WMMA f32)
// -------------------------------------------------------------------------
template <int CK>
__global__ __launch_bounds__(32)
void gemm64_kernel(const float* __restrict__ W, const float* __restrict__ h,
                   float* __restrict__ g, int C) {
    const int b  = blockIdx.y;
    const int n0 = blockIdx.x * 16;
    const int lane = threadIdx.x;
    const int hf = lane >> 4;
    const int lm = lane & 15;
    const float* hb = h + (size_t)b * C * NPTS;

    v8f acc[4] = {};
    #pragma unroll
    for (int kk = 0; kk < CK; ++kk) {
        int c0 = kk * 4 + hf * 2;
        v2f bfr;
        bfr.x = (c0     < C) ? hb[(c0    ) * NPTS + n0 + lm] : 0.0f;
        bfr.y = (c0 + 1 < C) ? hb[(c0 + 1) * NPTS + n0 + lm] : 0.0f;
        #pragma unroll
        for (int mt = 0; mt < 4; ++mt) {
            int o = mt * 16 + lm;
            v2f afr;
            afr.x = (c0     < C) ? W[o * C + c0    ] : 0.0f;
            afr.y = (c0 + 1 < C) ? W[o * C + c0 + 1] : 0.0f;
            acc[mt] = __builtin_amdgcn_wmma_f32_16x16x4_f32(
                false, afr, false, bfr, (short)0, acc[mt], false, false);
        }
    }
    float* gb = g + (size_t)b * 64 * NPTS;
    #pragma unroll
    for (int mt = 0; mt < 4; ++mt)
        #pragma unroll
        for (int r = 0; r < 8; ++r)
            gb[(mt * 16 + hf * 8 + r) * NPTS + n0 + lm] = acc[mt][r];
}

// -------------------------------------------------------------------------
// out[b,o,n] = relu(bn(max_k (g[b,o,idx[n,k]] - g[b,o,n])))
// bn affine a*x+b: max commutes -> pick max if a>=0 else min.
// -------------------------------------------------------------------------
__global__ void edge_max_kernel(const float* __restrict__ g,
                                const int* __restrict__ idx,
                                const float* __restrict__ bg, const float* __restrict__ bb,
                                const float* __restrict__ bm, const float* __restrict__ bv,
                                float* __restrict__ out) {
    const int b  = blockIdx.z;
    const int o  = threadIdx.x & 63;
    const int nl = threadIdx.x >> 6;
    const int n  = blockIdx.x * 4 + nl;
    const float* grow = g + ((size_t)b * 64 + o) * NPTS;
    const int* irow = idx + ((size_t)b * NPTS + n) * KNN;
    float gi = grow[n];
    float mx = -3.0e38f, mn = 3.0e38f;
    #pragma unroll 4
    for (int k = 0; k < KNN; ++k) {
        float d = grow[irow[k]] - gi;
        mx = fmaxf(mx, d); mn = fminf(mn, d);
    }
    float a  = bg[o] * rsqrtf(bv[o] + EPSBN);
    float bc = bb[o] - bm[o] * a;
    float m  = (a >= 0.f) ? mx : mn;
    out[((size_t)b * 64 + o) * NPTS + n] = fmaxf(a * m + bc, 0.f);
}

// -------------------------------------------------------------------------
// h5 = relu(bn5(W5 @ h)); fused max over N via shuffle + atomicMax.
// One wave per (ntile, mtile, b).  Post-ReLU values >= 0 -> int atomicMax ok.
// -------------------------------------------------------------------------
__global__ __launch_bounds__(32)
void gemm5_max_kernel(const float* __restrict__ W, const float* __restrict__ h,
                      const float* __restrict__ bg, const float* __restrict__ bb,
                      const float* __restrict__ bm, const float* __restrict__ bv,
                      float* __restrict__ hmax) {
    const int b  = blockIdx.z;
    const int m0 = blockIdx.y * 16;
    const int n0 = blockIdx.x * 16;
    const int lane = threadIdx.x;
    const int hf = lane >> 4;
    const int lm = lane & 15;
    const float* hb = h + (size_t)b * 64 * NPTS;

    v8f acc = {0.f,0.f,0.f,0.f,0.f,0.f,0.f,0.f};
    #pragma unroll
    for (int kk = 0; kk < 16; ++kk) {
        int c0 = kk * 4 + hf * 2;
        v2f bfr;
        bfr.x = hb[(c0    ) * NPTS + n0 + lm];
        bfr.y = hb[(c0 + 1) * NPTS + n0 + lm];
        v2f afr;
        afr.x = W[(m0 + lm) * 64 + c0    ];
        afr.y = W[(m0 + lm) * 64 + c0 + 1];
        acc = __builtin_amdgcn_wmma_f32_16x16x4_f32(
            false, afr, false, bfr, (short)0, acc, false, false);
    }
    float val[8];
    #pragma unroll
    for (int r = 0; r < 8; ++r) {
        int c = m0 + hf * 8 + r;
        float a = bg[c] * rsqrtf(bv[c] + EPSBN);
        val[r] = fmaxf((acc[r] - bm[c]) * a + bb[c], 0.f);
    }
    // max over the 16 columns of this tile (halves independent: masks < 16)
    #pragma unroll
    for (int r = 0; r < 8; ++r) {
        #pragma unroll
        for (int m = 1; m < 16; m <<= 1)
            val[r] = fmaxf(val[r], __shfl_xor(val[r], m, 32));
    }
    if (lm == 0) {
        #pragma unroll
        for (int r = 0; r < 8; ++r)
            atomicMax((int*)&hmax[(size_t)b * 1024 + m0 + hf * 8 + r],
                      __float_as_int(val[r]));
    }
}

__global__ void zero_kernel(float* __restrict__ p, int n) {
    int t = blockIdx.x * blockDim.x + threadIdx.x;
    if (t < n) p[t] = 0.f;
}

// -------------------------------------------------------------------------
// Fully connected: out[b,o] = act(bn?(in[b,:] @ W[o,:] + bias[o]))
// -------------------------------------------------------------------------
__global__ void fc_kernel(const float* __restrict__ in, int Cin,
                          const float* __restrict__ W, const float* __restrict__ bias,
                          const float* __restrict__ bg, const float* __restrict__ bb,
                          const float* __restrict__ bm, const float* __restrict__ bv,
                          float* __restrict__ out, int Cout, int useBN) {
    int t = blockIdx.x * blockDim.x + threadIdx.x;
    if (t >= NB * Cout) return;
    int b = t / Cout, o = t % Cout;
    const float* row = in + (size_t)b * Cin;
    const float* wr  = W + (size_t)o * Cin;
    float s = bias[o];
    for (int c = 0; c < Cin; ++c) s += row[c] * wr[c];
    if (useBN) {
        float a = bg[o] * rsqrtf(bv[o] + EPSBN);
        s = (s - bm[o]) * a + bb[o];
    }
    out[t] = fmaxf(s, 0.f);
}

// -------------------------------------------------------------------------
extern "C" void kernel_launch(void* const* d_in, const int* in_sizes, int n_in,
                              void* d_out, int out_size, void* d_ws, size_t ws_size,
                              hipStream_t stream) {
    (void)in_sizes; (void)n_in; (void)out_size; (void)ws_size;
    auto F = [&](int i) { return (const float*)d_in[i]; };

    // Input layout (setup_inputs dict order, flattened):
    // 0:x  1:w1 2-5:bn1  6:w2 7-10:bn2  11:w3 12-15:bn3  16:w4 17-20:bn4
    // 21:w5 22-25:bn5  26:w6 27:b6 28-31:bn6  32:w7 33:b7 34-37:bn7  38:w8 39:b8
    const float* x = F(0);

    float* ws   = (float*)d_ws;
    float* xxb  = ws;                  // 32768
    float* hmax = xxb  + 32768;        // 8192
    float* f6   = hmax + 8192;         // 4096
    float* f7   = f6   + 4096;         // 1024
    float* gbuf = f7   + 1024;         // 2097152
    float* hA   = gbuf + 2097152;      // 2097152
    float* hB   = hA   + 2097152;      // 2097152
    int*   idxb = (int*)(hB + 2097152);// 655360 ints

    dim3 gKnn(NPTS / 64, NB);
    dim3 g32(NPTS / 16, NB);
    auto run_block = [&](const float* hin, int C, int wI, int bnI, float* hout) {
        sqnorm_kernel<<<(NB * NPTS) / 256, 256, 0, stream>>>(hin, C, xxb);
        if (C == 3) {
            knn_kernel<1><<<gKnn, 128, 0, stream>>>(hin, xxb, idxb, C);
            gemm64_kernel<1><<<g32, 32, 0, stream>>>(F(wI), hin, gbuf, C);
        } else {
            knn_kernel<16><<<gKnn, 128, 0, stream>>>(hin, xxb, idxb, C);
            gemm64_kernel<16><<<g32, 32, 0, stream>>>(F(wI), hin, gbuf, C);
        }
        edge_max_kernel<<<dim3(NPTS / 4, 1, NB), 256, 0, stream>>>(
            gbuf, idxb, F(bnI), F(bnI + 1), F(bnI + 2), F(bnI + 3), hout);
    };

    run_block(x,  3,  1,  2, hA);
    run_block(hA, 64, 6,  7, hB);
    run_block(hB, 64, 11, 12, hA);
    run_block(hA, 64, 16, 17, hB);

    zero_kernel<<<(NB * 1024) / 256, 256, 0, stream>>>(hmax, NB * 1024);
    gemm5_max_kernel<<<dim3(NPTS / 16, 64, NB), 32, 0, stream>>>(
        F(21), hB, F(22), F(23), F(24), F(25), hmax);

    fc_kernel<<<(NB * 512 + 255) / 256, 256, 0, stream>>>(
        hmax, 1024, F(26), F(27), F(28), F(29), F(30), F(31), f6, 512, 1);
    fc_kernel<<<(NB * 128 + 255) / 256, 256, 0, stream>>>(
        f6, 512, F(32), F(33), F(34), F(35), F(36), F(37), f7, 128, 1);
    fc_kernel<<<(NB * 40 + 255) / 256, 256, 0, stream>>>(
        f7, 128, F(38), F(39), nullptr, nullptr, nullptr, nullptr,
        (float*)d_out, 40, 0);
}